// PhiLayer_635655159896
// MI455X (gfx1250) — compile-verified
//
#include <hip/hip_runtime.h>
#include <cstdint>

// ---------------------------------------------------------------------------
// Problem constants
// ---------------------------------------------------------------------------
#define B_   32
#define KV_  2048
#define HID_ 2560
#define NH_  32
#define HD_  80
#define RD_  32
#define NI_  10240
#define QKVW_ (3 * HID_)
#define LN_EPS 1e-5f
#define THETA_ 10000.0f

typedef float v2f __attribute__((ext_vector_type(2)));
typedef float v8f __attribute__((ext_vector_type(8)));
typedef int   v4i __attribute__((vector_size(16)));   // matches builtin param type

// ---------------------------------------------------------------------------
// gfx1250 async global->LDS path (guarded: falls back to plain LDS stores)
// ---------------------------------------------------------------------------
#if defined(__has_builtin)
#if __has_builtin(__builtin_amdgcn_global_load_async_to_lds_b128) && \
    __has_builtin(__builtin_amdgcn_s_wait_asynccnt)
#define USE_ASYNC_LDS 1
#endif
#endif
#ifndef USE_ASYNC_LDS
#define USE_ASYNC_LDS 0
#endif

__device__ __forceinline__ void cp_f4_to_lds(float* lds_dst, const float* g_src) {
#if USE_ASYNC_LDS
  __builtin_amdgcn_global_load_async_to_lds_b128(
      (__attribute__((address_space(1))) v4i*)g_src,
      (__attribute__((address_space(3))) v4i*)lds_dst,
      /*offset=*/0, /*cpol=*/0);
#else
  *reinterpret_cast<float4*>(lds_dst) = *reinterpret_cast<const float4*>(g_src);
#endif
}

__device__ __forceinline__ void cp_lds_wait() {
#if USE_ASYNC_LDS
  __builtin_amdgcn_s_wait_asynccnt(0);
#endif
}

// ---------------------------------------------------------------------------
// Kernel 1: LayerNorm over rows of [32, 2560]
// ---------------------------------------------------------------------------
__global__ void ln_kernel(const float* __restrict__ x, const float* __restrict__ g,
                          const float* __restrict__ bt, float* __restrict__ y) {
  const int row = blockIdx.x;
  const float* xr = x + (size_t)row * HID_;
  float s = 0.f, ss = 0.f;
  for (int i = threadIdx.x; i < HID_; i += 256) {
    float v = xr[i];
    s += v;
    ss += v * v;
  }
#pragma unroll
  for (int o = 16; o; o >>= 1) {
    s += __shfl_xor(s, o, 32);
    ss += __shfl_xor(ss, o, 32);
  }
  __shared__ float rs[8], rss[8];
  const int wave = threadIdx.x >> 5, lane = threadIdx.x & 31;
  if (lane == 0) { rs[wave] = s; rss[wave] = ss; }
  __syncthreads();
  if (wave == 0) {
    s = (lane < 8) ? rs[lane] : 0.f;
    ss = (lane < 8) ? rss[lane] : 0.f;
#pragma unroll
    for (int o = 4; o; o >>= 1) {
      s += __shfl_xor(s, o, 32);
      ss += __shfl_xor(ss, o, 32);
    }
    if (lane == 0) { rs[0] = s / (float)HID_; rss[0] = ss / (float)HID_; }
  }
  __syncthreads();
  const float mu = rs[0];
  const float var = rss[0] - mu * mu;
  const float rstd = rsqrtf(var + LN_EPS);
  for (int i = threadIdx.x; i < HID_; i += 256)
    y[(size_t)row * HID_ + i] = (xr[i] - mu) * rstd * g[i] + bt[i];
}

// ---------------------------------------------------------------------------
// Kernel 2 (generic): C[32,N] = A[32,K] @ W[N,K]^T + bias, fp32 WMMA 16x16x4.
//   epi: 0 = none, 1 = tanh-GELU, 2 = +add1 +add2 (residual fusion)
// Block = 128 threads (4 waves); each wave owns a 32x16 output strip.
// ---------------------------------------------------------------------------
__device__ __forceinline__ float gelu_tanh(float x) {
  const float k0 = 0.7978845608028654f;  // sqrt(2/pi)
  return 0.5f * x * (1.0f + tanhf(k0 * (x + 0.044715f * x * x * x)));
}

#define MT 32
#define KT 32
#define NT 64
#define LDSTRIDE (KT + 4)  // 36 floats -> 144B rows, 16B-aligned f4 slots

__global__ void gemm_wmma(const float* __restrict__ A, const float* __restrict__ W,
                          const float* __restrict__ bias, float* __restrict__ C,
                          int K, int N, int epi,
                          const float* __restrict__ add1,
                          const float* __restrict__ add2) {
  __shared__ __align__(16) float Alds[MT][LDSTRIDE];
  __shared__ __align__(16) float Wlds[NT][LDSTRIDE];

  const int tid = threadIdx.x;
  const int lane = tid & 31;
  const int wave = tid >> 5;
  const int n0 = blockIdx.x * NT;

  v8f acc0 = {0.f, 0.f, 0.f, 0.f, 0.f, 0.f, 0.f, 0.f};  // rows 0..15
  v8f acc1 = {0.f, 0.f, 0.f, 0.f, 0.f, 0.f, 0.f, 0.f};  // rows 16..31

  for (int k0 = 0; k0 < K; k0 += KT) {
    __syncthreads();  // LDS reuse guard
    // Stage A tile (32 x 32 floats) and W tile (64 x 32 floats), 128-bit chunks
#pragma unroll
    for (int i = tid; i < MT * (KT / 4); i += 128) {
      const int r = i >> 3, c4 = i & 7;
      cp_f4_to_lds(&Alds[r][c4 * 4], &A[(size_t)r * K + k0 + c4 * 4]);
    }
#pragma unroll
    for (int i = tid; i < NT * (KT / 4); i += 128) {
      const int r = i >> 3, c4 = i & 7;
      cp_f4_to_lds(&Wlds[r][c4 * 4], &W[(size_t)(n0 + r) * K + k0 + c4 * 4]);
    }
    if (k0 + KT < K) {  // speculative prefetch of next weight tile -> global_prefetch
      const float* pf = &W[(size_t)(n0 + (tid >> 1)) * K + k0 + KT + (tid & 1) * 16];
      __builtin_prefetch(pf, 0, 1);
    }
    cp_lds_wait();
    __syncthreads();

    // fp32 WMMA fragment layouts (16x4 A / 4x16 B, per CDNA5 ISA):
    //   A[m][k]: lane = m + 16*(k>>1), reg = k&1   (m = lane&15, khalf = lane>>4)
    const int khalf = lane >> 4;
    const int mr = lane & 15;
    const int nn = wave * 16 + mr;
#pragma unroll
    for (int kk = 0; kk < KT; kk += 4) {
      const int kb = kk + khalf * 2;
      v2f a0 = {Alds[mr][kb], Alds[mr][kb + 1]};
      v2f a1 = {Alds[16 + mr][kb], Alds[16 + mr][kb + 1]};
      v2f bb = {Wlds[nn][kb], Wlds[nn][kb + 1]};
      acc0 = __builtin_amdgcn_wmma_f32_16x16x4_f32(false, a0, false, bb, (short)0,
                                                   acc0, false, false);
      acc1 = __builtin_amdgcn_wmma_f32_16x16x4_f32(false, a1, false, bb, (short)0,
                                                   acc1, false, false);
    }
  }

  // Epilogue: C/D layout VGPR r -> M = r (lanes 0-15) / r+8 (lanes 16-31)
  const int nc = n0 + wave * 16 + (lane & 15);
  const int mbase = (lane < 16) ? 0 : 8;
  const float bv = bias[nc];
#pragma unroll
  for (int r = 0; r < 8; r++) {
    const int m = mbase + r;
    float v0 = acc0[r] + bv;
    float v1 = acc1[r] + bv;
    if (epi == 1) {
      v0 = gelu_tanh(v0);
      v1 = gelu_tanh(v1);
    } else if (epi == 2) {
      v0 += add1[(size_t)m * N + nc] + add2[(size_t)m * N + nc];
      v1 += add1[(size_t)(m + 16) * N + nc] + add2[(size_t)(m + 16) * N + nc];
    }
    C[(size_t)m * N + nc] = v0;
    C[(size_t)(m + 16) * N + nc] = v1;
  }
}

// ---------------------------------------------------------------------------
// Kernel 3: partial RoPE on q/k (first RD=32 dims of each 80-dim head)
// ---------------------------------------------------------------------------
__global__ void rope_kernel(const float* __restrict__ qkv, const int* __restrict__ lens,
                            float* __restrict__ qr, float* __restrict__ kr) {
  const int b = blockIdx.x;
  const float pos = (float)lens[b];
  const float* qs = qkv + (size_t)b * QKVW_;
  const float* ks = qs + HID_;
  for (int idx = threadIdx.x; idx < NH_ * HD_; idx += blockDim.x) {
    const int h = idx / HD_, d = idx % HD_;
    float qv, kv;
    if (d < RD_) {
      const int half = RD_ / 2;
      const int i = (d < half) ? d : d - half;
      const float freq = __powf(THETA_, -(float)(2 * i) / (float)RD_);
      float sn, cs;
      __sincosf(pos * freq, &sn, &cs);
      const float q1 = qs[h * HD_ + i], q2 = qs[h * HD_ + i + half];
      const float k1 = ks[h * HD_ + i], k2 = ks[h * HD_ + i + half];
      if (d < half) {
        qv = q1 * cs - q2 * sn;
        kv = k1 * cs - k2 * sn;
      } else {
        qv = q2 * cs + q1 * sn;
        kv = k2 * cs + k1 * sn;
      }
    } else {
      qv = qs[h * HD_ + d];
      kv = ks[h * HD_ + d];
    }
    qr[(size_t)b * HID_ + idx] = qv;
    kr[(size_t)b * HID_ + idx] = kv;
  }
}

// ---------------------------------------------------------------------------
// Kernel 4: decode attention. One block per (h, b). 8 waves.
// Position s == ctx uses the freshly-roped k / raw v (inputs stay immutable).
// ---------------------------------------------------------------------------
__global__ void attn_kernel(const float* __restrict__ qr, const float* __restrict__ kr,
                            const float* __restrict__ qkv,  // v lives at +2*HID per row
                            const float* __restrict__ kc, const float* __restrict__ vc,
                            const int* __restrict__ lens, float* __restrict__ out) {
  const int h = blockIdx.x, b = blockIdx.y;
  const int tid = threadIdx.x, lane = tid & 31, wave = tid >> 5;
  const int ctx = lens[b];  // valid positions: 0..ctx inclusive

  __shared__ float sc[KV_];
  __shared__ float red[8];
  __shared__ float oacc[HD_];
  if (tid < HD_) oacc[tid] = 0.f;

  const float* q = qr + ((size_t)b * NH_ + h) * HD_;
  const float q0 = q[lane];
  const float q1 = q[lane + 32];
  const float q2 = (lane < 16) ? q[lane + 64] : 0.f;
  const float scale = 0.11180339887498949f;  // 80^-0.5

  const float* knew = kr + ((size_t)b * NH_ + h) * HD_;
  for (int s = wave; s <= ctx; s += 8) {
    const float* kp =
        (s == ctx) ? knew : (kc + (((size_t)b * KV_ + s) * NH_ + h) * HD_);
    float d = q0 * kp[lane] + q1 * kp[lane + 32];
    if (lane < 16) d += q2 * kp[lane + 64];
#pragma unroll
    for (int o = 16; o; o >>= 1) d += __shfl_xor(d, o, 32);
    if (lane == 0) sc[s] = d * scale;
  }
  __syncthreads();

  // softmax over sc[0..ctx]
  float mx = -3.4e38f;
  for (int i = tid; i <= ctx; i += 256) mx = fmaxf(mx, sc[i]);
#pragma unroll
  for (int o = 16; o; o >>= 1) mx = fmaxf(mx, __shfl_xor(mx, o, 32));
  if (lane == 0) red[wave] = mx;
  __syncthreads();
  mx = red[0];
#pragma unroll
  for (int w = 1; w < 8; w++) mx = fmaxf(mx, red[w]);
  float sum = 0.f;
  for (int i = tid; i <= ctx; i += 256) {
    const float e = __expf(sc[i] - mx);
    sc[i] = e;
    sum += e;
  }
#pragma unroll
  for (int o = 16; o; o >>= 1) sum += __shfl_xor(sum, o, 32);
  __syncthreads();  // all reads of red (max) complete before reuse
  if (lane == 0) red[wave] = sum;
  __syncthreads();
  sum = 0.f;
#pragma unroll
  for (int w = 0; w < 8; w++) sum += red[w];
  const float inv = 1.0f / sum;

  // weighted V sum; lane covers d = lane, lane+32, lane+64(<80)
  float a0 = 0.f, a1 = 0.f, a2 = 0.f;
  const float* vnew = qkv + (size_t)b * QKVW_ + 2 * HID_ + h * HD_;
  for (int s = wave; s <= ctx; s += 8) {
    const float* vp =
        (s == ctx) ? vnew : (vc + (((size_t)b * KV_ + s) * NH_ + h) * HD_);
    const float p = sc[s];
    a0 += p * vp[lane];
    a1 += p * vp[lane + 32];
    if (lane < 16) a2 += p * vp[lane + 64];
  }
  atomicAdd(&oacc[lane], a0 * inv);          // ds_add_f32
  atomicAdd(&oacc[lane + 32], a1 * inv);
  if (lane < 16) atomicAdd(&oacc[lane + 64], a2 * inv);
  __syncthreads();
  if (tid < HD_) out[(size_t)b * HID_ + h * HD_ + tid] = oacc[tid];
}

// ---------------------------------------------------------------------------
// Launch
// ---------------------------------------------------------------------------
extern "C" void kernel_launch(void* const* d_in, const int* in_sizes, int n_in,
                              void* d_out, int out_size, void* d_ws, size_t ws_size,
                              hipStream_t stream) {
  (void)in_sizes; (void)n_in; (void)out_size; (void)ws_size;
  const float* hs    = (const float*)d_in[0];
  const int*   lens  = (const int*)d_in[1];
  const float* kc    = (const float*)d_in[2];
  const float* vc    = (const float*)d_in[3];
  const float* ln_g  = (const float*)d_in[4];
  const float* ln_b  = (const float*)d_in[5];
  const float* w_qkv = (const float*)d_in[6];
  const float* b_qkv = (const float*)d_in[7];
  const float* w_out = (const float*)d_in[8];
  const float* b_out = (const float*)d_in[9];
  const float* w_fc1 = (const float*)d_in[10];
  const float* b_fc1 = (const float*)d_in[11];
  const float* w_fc2 = (const float*)d_in[12];
  const float* b_fc2 = (const float*)d_in[13];
  float* out = (float*)d_out;

  float* ws    = (float*)d_ws;
  float* xn    = ws;                       // 32*2560
  float* qkv   = xn + (size_t)B_ * HID_;   // 32*7680
  float* q_r   = qkv + (size_t)B_ * QKVW_; // 32*2560
  float* k_r   = q_r + (size_t)B_ * HID_;  // 32*2560
  float* attn  = k_r + (size_t)B_ * HID_;  // 32*2560
  float* attnp = attn + (size_t)B_ * HID_; // 32*2560
  float* hmid  = attnp + (size_t)B_ * HID_;// 32*10240

  ln_kernel<<<B_, 256, 0, stream>>>(hs, ln_g, ln_b, xn);
  gemm_wmma<<<QKVW_ / NT, 128, 0, stream>>>(xn, w_qkv, b_qkv, qkv, HID_, QKVW_, 0,
                                            nullptr, nullptr);
  rope_kernel<<<B_, 256, 0, stream>>>(qkv, lens, q_r, k_r);
  attn_kernel<<<dim3(NH_, B_), 256, 0, stream>>>(q_r, k_r, qkv, kc, vc, lens, attn);
  gemm_wmma<<<HID_ / NT, 128, 0, stream>>>(attn, w_out, b_out, attnp, HID_, HID_, 0,
                                           nullptr, nullptr);
  gemm_wmma<<<NI_ / NT, 128, 0, stream>>>(xn, w_fc1, b_fc1, hmid, HID_, NI_, 1,
                                          nullptr, nullptr);
  gemm_wmma<<<HID_ / NT, 128, 0, stream>>>(hmid, w_fc2, b_fc2, out, NI_, HID_, 2,
                                           hs, attnp);
}